// ScaledDPAttention_54881092108990
// MI455X (gfx1250) — compile-verified
//
#include <hip/hip_runtime.h>
#include <math.h>

// ---------------------------------------------------------------------------
// ScaledDPAttention (algebraically collapsed attention) for MI455X / gfx1250.
// B=16, L=2048, D=256, NCLASS=1000.
// Memory-bound on the 3 embedding gathers (~100MB -> ~4.3us @ 23.3TB/s).
// Decoder GEMM uses V_WMMA_F32_16X16X4_F32 (full f32 precision).
// ---------------------------------------------------------------------------

#define BB      16
#define LL      2048
#define DD      256
#define NCLS    1000
#define LN_EPS  1e-5f

typedef __attribute__((ext_vector_type(2))) float v2f;
typedef __attribute__((ext_vector_type(8))) float v8f;

// ---- workspace layout (float offsets) ----
#define WS_KSUM   0                 // [16*256]
#define WS_VSUM   4096              // [16*256]
#define WS_OUTACC 8192              // [16*256]
#define WS_DENOM  12288             // [16]
#define WS_LENS   12304             // [16]
#define WS_HNORM  12320             // [16*256]
#define WS_Y      16416             // [16*2048]
#define WS_ZERO_N 12320             // floats to zero (accumulators)

// ---------------------------------------------------------------------------
__global__ void sdpa_zero_kernel(float* __restrict__ w, int n) {
    int i = blockIdx.x * blockDim.x + threadIdx.x;
    if (i < n) w[i] = 0.0f;
}

// ---------------------------------------------------------------------------
// Pass 1: ksum[b,d] = sum_l mask[b,l]*Wk[ids[b,l],d] ; lens[b] = sum_l mask.
// 256 threads = one dim each; 16 chunks per batch; atomic f32 reduction.
__global__ void sdpa_ksum_kernel(const int* __restrict__ ids,
                                 const float* __restrict__ mask,
                                 const float* __restrict__ Wk,
                                 float* __restrict__ ksum,
                                 float* __restrict__ lens) {
    const int CH = 16, TPB = LL / CH;          // 128 tokens / block
    int b = blockIdx.x / CH;
    int l0 = (blockIdx.x % CH) * TPB;
    int t = threadIdx.x;                       // dim 0..255
    float acc = 0.0f, msum = 0.0f;
    for (int l = l0; l < l0 + TPB; ++l) {
        int id = ids[b * LL + l];
        float m = mask[b * LL + l];
        __builtin_prefetch(ids + b * LL + l + 8, 0, 0);
        acc  += m * Wk[(size_t)id * DD + t];
        msum += m;
    }
    atomicAdd(&ksum[b * DD + t], acc);
    if (t == 0) atomicAdd(&lens[b], msum);
}

// ---------------------------------------------------------------------------
// Pass 2: s[b,l] = dot(Wq[ids]*m, ksum[b]) / 16 ; y = exp(s)*m ; denom[b]=sum y.
// One wave per token (lanes split the 256-dim dot, shuffle-reduce).
__global__ void sdpa_score_kernel(const int* __restrict__ ids,
                                  const float* __restrict__ mask,
                                  const float* __restrict__ Wq,
                                  const float* __restrict__ ksum,
                                  float* __restrict__ y,
                                  float* __restrict__ denom) {
    const int CH = 8, TPB = LL / CH;           // 256 tokens / block, 8 waves
    int b = blockIdx.x / CH;
    int l0 = (blockIdx.x % CH) * TPB;
    int t = threadIdx.x;
    __shared__ float ks[DD];
    ks[t] = ksum[b * DD + t];
    __syncthreads();
    int lane = t & 31, wave = t >> 5;
    float dsum = 0.0f;
    for (int l = l0 + wave; l < l0 + TPB; l += 8) {
        int id = ids[b * LL + l];
        const float4* qr = (const float4*)(Wq + (size_t)id * DD + lane * 8);
        float4 q0 = qr[0], q1 = qr[1];
        float p = q0.x * ks[lane * 8 + 0] + q0.y * ks[lane * 8 + 1] +
                  q0.z * ks[lane * 8 + 2] + q0.w * ks[lane * 8 + 3] +
                  q1.x * ks[lane * 8 + 4] + q1.y * ks[lane * 8 + 5] +
                  q1.z * ks[lane * 8 + 6] + q1.w * ks[lane * 8 + 7];
        #pragma unroll
        for (int off = 16; off > 0; off >>= 1) p += __shfl_xor(p, off, 32);
        if (lane == 0) {
            float m = mask[b * LL + l];
            float yy = __expf(p * 0.0625f) * m;   // 1/sqrt(256) = 1/16
            y[b * LL + l] = yy;
            dsum += yy;
        }
    }
    if (lane == 0) atomicAdd(&denom[b], dsum);
}

// ---------------------------------------------------------------------------
// Pass 3: outacc[b,d] = sum_l Wv[id,d]*m*y ; vsum[b,d] = sum_l Wv[id,d]*m.
__global__ void sdpa_vacc_kernel(const int* __restrict__ ids,
                                 const float* __restrict__ mask,
                                 const float* __restrict__ Wv,
                                 const float* __restrict__ y,
                                 float* __restrict__ outacc,
                                 float* __restrict__ vsum) {
    const int CH = 16, TPB = LL / CH;
    int b = blockIdx.x / CH;
    int l0 = (blockIdx.x % CH) * TPB;
    int t = threadIdx.x;
    float ao = 0.0f, as = 0.0f;
    for (int l = l0; l < l0 + TPB; ++l) {
        int id = ids[b * LL + l];
        float m = mask[b * LL + l];
        float yy = y[b * LL + l];
        float v = Wv[(size_t)id * DD + t];
        ao += v * (m * yy);
        as += v * m;
    }
    atomicAdd(&outacc[b * DD + t], ao);
    atomicAdd(&vsum[b * DD + t], as);
}

// ---------------------------------------------------------------------------
// Pass 4: h = outacc/denom + vsum/lens ; LayerNorm over d -> hnorm.
__global__ void sdpa_ln_kernel(const float* __restrict__ outacc,
                               const float* __restrict__ vsum,
                               const float* __restrict__ denom,
                               const float* __restrict__ lens,
                               float* __restrict__ hnorm) {
    int b = blockIdx.x, t = threadIdx.x;
    __shared__ float red[DD];
    float h = outacc[b * DD + t] / denom[b] + vsum[b * DD + t] / lens[b];
    red[t] = h;
    __syncthreads();
    for (int s = DD / 2; s > 0; s >>= 1) {
        if (t < s) red[t] += red[t + s];
        __syncthreads();
    }
    float mu = red[0] * (1.0f / DD);
    __syncthreads();
    float d = h - mu;
    red[t] = d * d;
    __syncthreads();
    for (int s = DD / 2; s > 0; s >>= 1) {
        if (t < s) red[t] += red[t + s];
        __syncthreads();
    }
    float var = red[0] * (1.0f / DD);
    hnorm[b * DD + t] = d * rsqrtf(var + LN_EPS);
}

// ---------------------------------------------------------------------------
// Pass 5: out[16,1000] = hnorm[16,256] @ Wdec^T, via V_WMMA_F32_16X16X4_F32.
// One wave per 16x16 output tile; 64 WMMA accumulation steps (K=256, 4/step).
// A layout (16x4 f32): lane = m + 16*(k/2), reg = k%2 -> per-lane float2 load.
// B layout (4x16 f32): n = lane&15, k = 2*(lane>>4) + reg -> per-lane float2.
// C/D layout (16x16 f32, v8f): row = reg + 8*(lane>>4), col = lane&15.
// Tail tile (classes 992..1007) handled branchlessly: clamp the B row index
// into range so loads are always legal, then select-zero the fragment for
// out-of-range classes (keeps EXEC all-ones, no saveexec around loads).
__global__ void sdpa_decode_wmma_kernel(const float* __restrict__ hnorm,
                                        const float* __restrict__ Wdec,
                                        float* __restrict__ out) {
    int c0   = blockIdx.x * 16;        // class tile base (0..62)
    int lane = threadIdx.x;            // 0..31 (wave32)
    int m    = lane & 15;
    int grp  = lane >> 4;              // selects K pair / row half
    int cls  = c0 + m;                 // class row feeding B fragment
    float bsel = (cls < NCLS) ? 1.0f : 0.0f;
    int clsc = (cls < NCLS) ? cls : (NCLS - 1);
    const float* brow = Wdec + (size_t)clsc * DD;
    const float* arow = hnorm + (size_t)m * DD;

    v8f acc = {};
    #pragma unroll 4
    for (int k0 = 0; k0 < DD; k0 += 4) {
        int ka = k0 + 2 * grp;
        v2f a  = *(const v2f*)(arow + ka);
        v2f bb = *(const v2f*)(brow + ka);
        bb[0] *= bsel;                 // v_cndmask/v_mul, no exec branching
        bb[1] *= bsel;
        acc = __builtin_amdgcn_wmma_f32_16x16x4_f32(
            /*neg_a=*/false, a, /*neg_b=*/false, bb,
            /*c_mod=*/(short)0, acc, /*reuse_a=*/false, /*reuse_b=*/false);
    }

    int col = c0 + (lane & 15);
    if (col < NCLS) {
        #pragma unroll
        for (int r = 0; r < 8; ++r) {
            int row = r + 8 * grp;
            out[row * NCLS + col] = acc[r];
        }
    }
}

// ---------------------------------------------------------------------------
extern "C" void kernel_launch(void* const* d_in, const int* in_sizes, int n_in,
                              void* d_out, int out_size, void* d_ws, size_t ws_size,
                              hipStream_t stream) {
    const int*   ids  = (const int*)d_in[0];
    const float* mask = (const float*)d_in[1];
    const float* Wv   = (const float*)d_in[2];
    const float* Wk   = (const float*)d_in[3];
    const float* Wq   = (const float*)d_in[4];
    const float* Wdec = (const float*)d_in[5];
    float* out = (float*)d_out;
    float* w   = (float*)d_ws;

    float* ksum   = w + WS_KSUM;
    float* vsum   = w + WS_VSUM;
    float* outacc = w + WS_OUTACC;
    float* denom  = w + WS_DENOM;
    float* lens   = w + WS_LENS;
    float* hnorm  = w + WS_HNORM;
    float* y      = w + WS_Y;

    // 0) zero the atomic accumulators
    sdpa_zero_kernel<<<(WS_ZERO_N + 255) / 256, 256, 0, stream>>>(w, WS_ZERO_N);
    // 1) ksum + lens (Wk gather)
    sdpa_ksum_kernel<<<BB * 16, 256, 0, stream>>>(ids, mask, Wk, ksum, lens);
    // 2) scores -> y, denom (Wq gather)
    sdpa_score_kernel<<<BB * 8, 256, 0, stream>>>(ids, mask, Wq, ksum, y, denom);
    // 3) weighted V sums (Wv gather)
    sdpa_vacc_kernel<<<BB * 16, 256, 0, stream>>>(ids, mask, Wv, y, outacc, vsum);
    // 4) combine + LayerNorm
    sdpa_ln_kernel<<<BB, 256, 0, stream>>>(outacc, vsum, denom, lens, hnorm);
    // 5) decoder GEMM via f32 WMMA: 63 tiles of 16 classes, one wave each
    sdpa_decode_wmma_kernel<<<(NCLS + 15) / 16, 32, 0, stream>>>(hnorm, Wdec, out);
}